// AnatomicalContrastiveLoss_17910013624794
// MI455X (gfx1250) — compile-verified
//
#include <hip/hip_runtime.h>
#include <math.h>

typedef __attribute__((ext_vector_type(2))) float v2f;
typedef __attribute__((ext_vector_type(8))) float v8f;

#define N_CLASSES 16
#define N_FEAT    64
#define BATCH     4
#define TOPK      100
#define NV        262144           // 2^18
#define TAU_INV   10.0f
#define THETA     0.9f

// K2 tiling: 64 blocks/batch, 4096 voxels/block, LDS tiles of 128 voxels.
#define TILE_V    128
#define LDS_STRIDE 132             // 132 mod 64 == 4 -> conflict-free b64 reads
#define TILES_PER_BLOCK 32         // 4096 / 128

// ---------------- K0: zero the small accumulators ----------------
__global__ void k0_init(int* counts, float* sums, float* out) {
    int t = threadIdx.x;
    if (t < N_CLASSES) counts[t] = 0;
    for (int i = t; i < N_CLASSES * N_FEAT; i += 256) sums[i] = 0.f;
    if (t == 0) out[0] = 0.f;
}

// ---------------- K1: labels (argmax y), weights (prod proba), counts ----
// One streaming pass over y (64MB) + proba (64MB), coalesced along v.
__global__ __launch_bounds__(256) void k1_prep(const float* __restrict__ proba,
                                               const float* __restrict__ y,
                                               int* __restrict__ labels,
                                               float* __restrict__ weights,
                                               int* __restrict__ counts) {
    __shared__ int lc[N_CLASSES];
    if (threadIdx.x < N_CLASSES) lc[threadIdx.x] = 0;
    __syncthreads();

    int t = blockIdx.x * 256 + threadIdx.x;     // t in [0, B*V)
    int b = t >> 18;
    int v = t & (NV - 1);
    size_t base = (size_t)b * N_CLASSES * NV + v;

    float w = 1.f, besty = -1.f;
    int lab = 0;
#pragma unroll
    for (int c = 0; c < N_CLASSES; ++c) {
        w *= proba[base + (size_t)c * NV];
        float yv = y[base + (size_t)c * NV];
        if (yv > besty) { besty = yv; lab = c; }
    }
    labels[t]  = lab;
    weights[t] = w;
    atomicAdd(&lc[lab], 1);
    __syncthreads();
    if (threadIdx.x < N_CLASSES && lc[threadIdx.x] > 0)
        atomicAdd(&counts[threadIdx.x], lc[threadIdx.x]);
}

// ---------------- K2: one-hot GEMM  sums[c,f] += onehot * emb  via WMMA ---
// LDS-staged: coalesced global_load_b128 into a padded [64 x 132] tile,
// WMMA B operands fed by conflict-free ds_load_b64.
// A(16x4 f32) = onehot(class, voxel), B(4x16 f32) = emb(voxel, feature tile).
__global__ __launch_bounds__(256) void k2_sums_wmma(const float* __restrict__ emb,
                                                    const int* __restrict__ labels,
                                                    float* __restrict__ sums) {
    __shared__ float stile[N_FEAT * LDS_STRIDE];   // 33,792 B
    __shared__ int   ltile[TILE_V];
    __shared__ float ssum[N_CLASSES * N_FEAT];     // 4 KB

    for (int i = threadIdx.x; i < N_CLASSES * N_FEAT; i += 256) ssum[i] = 0.f;
    __syncthreads();

    const int b       = blockIdx.x >> 6;
    const int blkInB  = blockIdx.x & 63;
    const int waveIn  = threadIdx.x >> 5;
    const int lane    = threadIdx.x & 31;
    const int half    = lane >> 4;          // 0: K=0,1   1: K=2,3
    const int n       = lane & 15;          // A: class row  |  B/D: column

    const int*   labBase = labels + b * NV;
    const float* embBase = emb + (size_t)b * N_FEAT * NV;
    const int    vBlock  = blkInB * (TILE_V * TILES_PER_BLOCK);   // 4096/block

    v8f acc0 = {}, acc1 = {}, acc2 = {}, acc3 = {};

    for (int ti = 0; ti < TILES_PER_BLOCK; ++ti) {
        const int v0 = vBlock + ti * TILE_V;

        // --- stage: 64 rows x 128 voxels, 512B contiguous per wave per row ---
        for (int i = threadIdx.x; i < N_FEAT * (TILE_V / 4); i += 256) {
            int f = i >> 5;                  // TILE_V/4 = 32 float4 per row
            int pos = i & 31;
            float4 d = *(const float4*)(embBase + (size_t)f * NV + v0 + 4 * pos);
            *(float4*)&stile[f * LDS_STRIDE + 4 * pos] = d;
        }
        if (threadIdx.x < TILE_V) ltile[threadIdx.x] = labBase[v0 + threadIdx.x];
        __syncthreads();

        // --- compute: 32 chunks of 4 voxels, 4 per wave ---
#pragma unroll
        for (int cc = 0; cc < 4; ++cc) {
            const int chunk = waveIn * 4 + cc;          // 0..31
            const int vk    = chunk * 4 + 2 * half;     // this lane's voxel pair
            const int l0 = ltile[vk];
            const int l1 = ltile[vk + 1];
            v2f a;
            a.x = (l0 == n) ? 1.f : 0.f;                // A[n, 2*half]
            a.y = (l1 == n) ? 1.f : 0.f;                // A[n, 2*half+1]
            v2f b0 = *(const v2f*)&stile[( 0 + n) * LDS_STRIDE + vk];
            v2f b1 = *(const v2f*)&stile[(16 + n) * LDS_STRIDE + vk];
            v2f b2 = *(const v2f*)&stile[(32 + n) * LDS_STRIDE + vk];
            v2f b3 = *(const v2f*)&stile[(48 + n) * LDS_STRIDE + vk];
            acc0 = __builtin_amdgcn_wmma_f32_16x16x4_f32(false, a, false, b0, (short)0, acc0, false, false);
            acc1 = __builtin_amdgcn_wmma_f32_16x16x4_f32(false, a, false, b1, (short)0, acc1, false, false);
            acc2 = __builtin_amdgcn_wmma_f32_16x16x4_f32(false, a, false, b2, (short)0, acc2, false, false);
            acc3 = __builtin_amdgcn_wmma_f32_16x16x4_f32(false, a, false, b3, (short)0, acc3, false, false);
        }
        __syncthreads();
    }

    // D layout: VGPR r -> M = r + 8*half, N = n. Reduce across 8 waves in LDS.
#pragma unroll
    for (int r = 0; r < 8; ++r) {
        int m = r + 8 * half;
        atomicAdd(&ssum[m * N_FEAT +  0 + n], acc0[r]);
        atomicAdd(&ssum[m * N_FEAT + 16 + n], acc1[r]);
        atomicAdd(&ssum[m * N_FEAT + 32 + n], acc2[r]);
        atomicAdd(&ssum[m * N_FEAT + 48 + n], acc3[r]);
    }
    __syncthreads();
    for (int i = threadIdx.x; i < N_CLASSES * N_FEAT; i += 256)
        atomicAdd(&sums[i], ssum[i]);
}

// ---------------- K3: per-batch top-100 by iterative argmax ---------------
// Deterministic; ties -> lowest index (matches jax.lax.top_k). weights is
// rewritten by K1 each call, so destroying entries here is replay-safe.
__global__ __launch_bounds__(512) void k3_topk(float* __restrict__ weights,
                                               const int* __restrict__ labels,
                                               int* __restrict__ idx,
                                               int* __restrict__ hec) {
    const int b = blockIdx.x;
    float* w = weights + b * NV;
    __shared__ float sv[512];
    __shared__ int   si[512];

    for (int iter = 0; iter < TOPK; ++iter) {
        float best = -1.f; int bi = 0;
        for (int i = threadIdx.x; i < NV; i += 512) {
            float x = w[i];
            if (x > best) { best = x; bi = i; }
        }
        sv[threadIdx.x] = best; si[threadIdx.x] = bi;
        __syncthreads();
        for (int s = 256; s > 0; s >>= 1) {
            if (threadIdx.x < s) {
                float ov = sv[threadIdx.x + s]; int oi = si[threadIdx.x + s];
                if (ov > sv[threadIdx.x] ||
                    (ov == sv[threadIdx.x] && oi < si[threadIdx.x])) {
                    sv[threadIdx.x] = ov; si[threadIdx.x] = oi;
                }
            }
            __syncthreads();
        }
        if (threadIdx.x == 0) {
            int sel = si[0];
            idx[b * TOPK + iter] = sel;
            hec[b * TOPK + iter] = labels[b * NV + sel];
            w[sel] = -1.f;                       // exclude from later rounds
        }
        __syncthreads();
    }
}

// ---------------- K4a: avg = theta * sums / count (count>0) ----------------
__global__ void k4a_avg(const float* __restrict__ sums,
                        const int* __restrict__ counts,
                        float* __restrict__ avg) {
    int id = blockIdx.x * 256 + threadIdx.x;     // < 1024
    if (id >= N_CLASSES * N_FEAT) return;
    int c = id >> 6;
    int cnt = counts[c];
    avg[id] = (cnt > 0) ? THETA * sums[id] / (float)cnt : 0.f;
}

// ---------------- K4b: per (b,k): gather he, compute S, E_myc, L_myc ------
__global__ __launch_bounds__(64) void k4b_prep(const float* __restrict__ emb,
                                               const int* __restrict__ idx,
                                               const int* __restrict__ hec,
                                               const float* __restrict__ avg,
                                               float* __restrict__ Ek,
                                               float* __restrict__ Lk,
                                               float* __restrict__ Sk) {
    int b = blockIdx.x / TOPK;
    int k = blockIdx.x % TOPK;
    int f = threadIdx.x;
    int vi  = idx[b * TOPK + k];
    int myc = hec[b * TOPK + k];
    float hev = emb[(size_t)(b * N_FEAT + f) * NV + vi];

    float S = 0.f, Ec = 0.f, Lc = 0.f;
#pragma unroll
    for (int c = 0; c < N_CLASSES; ++c) {
        float lg = hev * avg[c * N_FEAT + f] * TAU_INV;
        float e  = expf(lg);
        S += e;
        if (c == myc) { Ec = e; Lc = lg; }
    }
    int o = (b * TOPK + k) * N_FEAT + f;
    Ek[o] = Ec; Lk[o] = Lc; Sk[o] = S;
}

// ---------------- K4c: same-class pair loss, accumulate scalar ------------
__global__ __launch_bounds__(256) void k4c_pairs(const int* __restrict__ hec,
                                                 const float* __restrict__ Ek,
                                                 const float* __restrict__ Lk,
                                                 const float* __restrict__ Sk,
                                                 float* __restrict__ out) {
    const int b = blockIdx.x >> 4;
    const int c = blockIdx.x & 15;
    __shared__ int list[TOPK];
    __shared__ int snk;
    __shared__ float red[256];

    if (threadIdx.x == 0) {
        int m = 0;
        for (int k = 0; k < TOPK; ++k)
            if (hec[b * TOPK + k] == c) list[m++] = k;
        snk = m;
    }
    __syncthreads();
    const int m = snk;
    if (m == 0) return;

    float local = 0.f;
    for (int p = threadIdx.x; p < m * m; p += 256) {
        int i = list[p / m];
        int j = list[p % m];
        const float* Ei = Ek + (size_t)(b * TOPK + i) * N_FEAT;
        const float* Li = Lk + (size_t)(b * TOPK + i) * N_FEAT;
        const float* Ej = Ek + (size_t)(b * TOPK + j) * N_FEAT;
        const float* Sj = Sk + (size_t)(b * TOPK + j) * N_FEAT;
        float s = 0.f;
        for (int f = 0; f < N_FEAT; ++f)
            s += logf(Ei[f] + Sj[f] - Ej[f]) - Li[f];
        local += s;
    }
    red[threadIdx.x] = local;
    __syncthreads();
    for (int s = 128; s > 0; s >>= 1) {
        if (threadIdx.x < s) red[threadIdx.x] += red[threadIdx.x + s];
        __syncthreads();
    }
    if (threadIdx.x == 0) {
        float outbc = red[0] / (float)(m * m * N_FEAT);
        atomicAdd(out, -outbc * (1.0f / BATCH));
    }
}

// ---------------- launch ----------------
extern "C" void kernel_launch(void* const* d_in, const int* in_sizes, int n_in,
                              void* d_out, int out_size, void* d_ws, size_t ws_size,
                              hipStream_t stream) {
    const float* proba = (const float*)d_in[0];
    const float* y     = (const float*)d_in[1];
    const float* emb   = (const float*)d_in[2];
    float* out = (float*)d_out;

    char* ws = (char*)d_ws;
    const size_t MB4 = (size_t)4 << 20;
    int*   labels  = (int*)  (ws);                       // B*V ints   (4MB)
    float* weights = (float*)(ws + MB4);                 // B*V floats (4MB)
    char*  ws8     = ws + 2 * MB4;
    int*   counts  = (int*)  (ws8);                      // 16
    float* sums    = (float*)(ws8 + 512);                // 1024
    float* avg     = (float*)(ws8 + 512 + 4096);         // 1024
    int*   idx     = (int*)  (ws8 + 8704);               // B*K
    int*   hec     = (int*)  (ws8 + 10752);              // B*K
    float* Ek      = (float*)(ws8 + 12800);              // B*K*F
    float* Lk      = (float*)(ws8 + 12800 + 102400);
    float* Sk      = (float*)(ws8 + 12800 + 204800);

    k0_init   <<<1, 256, 0, stream>>>(counts, sums, out);
    k1_prep   <<<(BATCH * NV) / 256, 256, 0, stream>>>(proba, y, labels, weights, counts);
    k2_sums_wmma<<<BATCH * 64, 256, 0, stream>>>(emb, labels, sums);
    k3_topk   <<<BATCH, 512, 0, stream>>>(weights, labels, idx, hec);
    k4a_avg   <<<4, 256, 0, stream>>>(sums, counts, avg);
    k4b_prep  <<<BATCH * TOPK, 64, 0, stream>>>(emb, idx, hec, avg, Ek, Lk, Sk);
    k4c_pairs <<<BATCH * N_CLASSES, 256, 0, stream>>>(hec, Ek, Lk, Sk, out);
}